// SSWPredictorLSTM_64166811402870
// MI455X (gfx1250) — compile-verified
//
#include <hip/hip_runtime.h>
#include <hip/hip_bf16.h>
#include <math.h>

// ---------------------------------------------------------------------------
// SSWPredictorLSTM for MI455X (gfx1250)
//   Phase 1: gates_x = seq @ W_ih^T + (b_ih+b_hh)   -> v_wmma_f32_16x16x4_f32
//   Phase 2: LSTM scan, W_hh partitioned across 16 WGPs' 320KB LDS
//            (staged via Tensor Data Mover), h exchanged via L2 +
//            device-scope barrier per step, next-step gates prefetched
//   Phase 3: log_softmax over width-1 axis == exactly 0 -> write zeros
// ---------------------------------------------------------------------------

typedef __attribute__((ext_vector_type(2))) float    v2f;
typedef __attribute__((ext_vector_type(8))) float    v8f;
typedef __attribute__((ext_vector_type(4))) unsigned u32x4;
typedef __attribute__((ext_vector_type(4))) int      i32x4;
typedef __attribute__((ext_vector_type(8))) int      i32x8;

#define T_SEQ 65536
#define D_IN  128
#define H_DIM 512
#define G4H   2048   // 4*H

#ifndef __has_builtin
#define __has_builtin(x) 0
#endif
#if __has_builtin(__builtin_amdgcn_tensor_load_to_lds) && \
    __has_builtin(__builtin_amdgcn_s_wait_tensorcnt)
#define HAVE_TDM 1
#else
#define HAVE_TDM 0
#endif

// d_ws layout (bytes):
//   [0,    2048) h state   (512 f32)
//   [2048, 4096) c state   (512 f32)
//   [4096, 4104) barrier {count, generation}
//   [8192, ... ) gates_x chunk buffer (chunkT x 2048 f32)
#define WS_GATES_OFF 8192

// -------------------------- init: zero state region ------------------------
__global__ void lstm_init_state(unsigned* ws) {
    for (int i = threadIdx.x; i < 2048; i += 256) ws[i] = 0u;
}

// ------------------ Phase 1: gates_x GEMM via f32 WMMA ---------------------
// Block: 256 threads = 8 waves, arranged 2(M) x 4(N); block tile 32 x 64.
// Each wave computes one 16x16 tile with 32 chained V_WMMA_F32_16X16X4_F32.
__global__ void lstm_gatesx_wmma(const float* __restrict__ seq,
                                 const float* __restrict__ W_ih,
                                 const float* __restrict__ b_ih,
                                 const float* __restrict__ b_hh,
                                 float* __restrict__ gates,
                                 int c0)
{
    const int lane  = threadIdx.x & 31;
    const int wv    = threadIdx.x >> 5;     // 0..7
    const int half  = lane >> 4;            // 0/1
    const int idx   = lane & 15;            // 0..15
    const int m_sub = wv & 1;
    const int n_sub = wv >> 1;

    const int t0 = c0 + (int)blockIdx.y * 32 + m_sub * 16;   // timestep tile
    const int j0 = (int)blockIdx.x * 64 + n_sub * 16;        // gate-dim tile

    // EXEC must stay all-ones across WMMA -> clamp addresses, no branches.
    int ta = t0 + idx; if (ta > T_SEQ - 1) ta = T_SEQ - 1;
    const int jb = j0 + idx;                                  // B col = W_ih row

    const float* __restrict__ arow = seq  + (size_t)ta * D_IN;
    const float* __restrict__ brow = W_ih + (size_t)jb * D_IN; // B[k][n]=W_ih[j,k]

    v8f acc = {};
    #pragma unroll
    for (int kb = 0; kb < D_IN; kb += 4) {
        v2f a = *(const v2f*)(arow + kb + 2 * half);   // A[m, kb+2h .. +1]
        v2f b = *(const v2f*)(brow + kb + 2 * half);   // B[kb+2h .., n]
        acc = __builtin_amdgcn_wmma_f32_16x16x4_f32(
                  /*neg_a=*/false, a, /*neg_b=*/false, b,
                  /*c_mod=*/(short)0, acc, /*reuse_a=*/false, /*reuse_b=*/false);
    }

    const float bias = b_ih[jb] + b_hh[jb];
    #pragma unroll
    for (int r = 0; r < 8; ++r) {
        const int t = t0 + r + 8 * half;                // D: VGPR r -> M=r+8*half
        if (t < T_SEQ)
            gates[(size_t)(t - c0) * G4H + jb] = acc[r] + bias;
    }
}

// --------------------------- device-scope barrier --------------------------
__device__ inline void grid_barrier(unsigned* bar, unsigned nblocks) {
    __syncthreads();
    if (threadIdx.x == 0) {
        __threadfence();                       // release h stores device-wide
        volatile unsigned* vgen = bar + 1;
        const unsigned g = *vgen;
        if (atomicAdd(bar, 1u) == nblocks - 1u) {
            bar[0] = 0u;
            __threadfence();
            atomicAdd(bar + 1, 1u);            // flip generation
        } else {
            while (*vgen == g) __builtin_amdgcn_s_sleep(2);
        }
    }
    __syncthreads();
}

__device__ inline float sigmoidf_(float x) { return 1.0f / (1.0f + __expf(-x)); }

#if HAVE_TDM
// One 2D TDM descriptor: `rows` contiguous rows of `rowlen` f32 from global
// -> LDS at byte offset lds_byte_off. D# bitfields per CDNA5 ISA §8.3/§8.4.
__device__ inline void tdm_load_rows_f32(const float* gsrc, unsigned lds_byte_off,
                                         unsigned rows, unsigned rowlen) {
    const unsigned long long ga = (unsigned long long)(size_t)gsrc;
    u32x4 g0 = {0u, 0u, 0u, 0u};
    g0[0] = 1u;                                            // count=1 (valid user D#)
    g0[1] = lds_byte_off;                                  // lds_addr
    g0[2] = (unsigned)(ga & 0xFFFFFFFFull);                // global_addr[31:0]
    g0[3] = (unsigned)((ga >> 32) & 0x01FFFFFFull)         // global_addr[56:32]
          | (2u << 30);                                    // type=2 ("image")
    i32x8 g1 = {0, 0, 0, 0, 0, 0, 0, 0};
    g1[0] = (int)(2u << 16);                               // data_size=4B; no mask
    g1[1] = (int)(rowlen << 16);                           // tensor_dim0[15:0]
    g1[2] = (int)((rowlen >> 16) | (rows << 16));          // dim0 hi | tensor_dim1 lo
    g1[3] = (int)((rows >> 16) | (rowlen << 16));          // dim1 hi | tile_dim0
    g1[4] = (int)rows;                                     // tile_dim1 | tile_dim2=0
    g1[5] = (int)rowlen;                                   // tensor_dim0_stride lo32
    i32x4 g2 = {0, 0, 0, 0};                               // 2D tensor: groups 2/3
    i32x4 g3 = {0, 0, 0, 0};                               //   unused
#if __clang_major__ >= 23
    i32x8 g4 = {0, 0, 0, 0, 0, 0, 0, 0};
    __builtin_amdgcn_tensor_load_to_lds(g0, g1, g2, g3, g4, 0);
#else
    __builtin_amdgcn_tensor_load_to_lds(g0, g1, g2, g3, 0);
#endif
}
#endif

// ---------------------- Phase 2: LSTM recurrence scan ----------------------
// 16 blocks x 256 threads. Block b owns hidden units [b*32, b*32+32) and the
// 128 corresponding rows of W_hh (rows g*512 + b*32 + u) in LDS (256KB).
__global__ void lstm_scan(const float* __restrict__ W_hh,
                          const float* __restrict__ gates,
                          float* __restrict__ ws_state,
                          int c0, int steps, unsigned nblocks)
{
    extern __shared__ float smem[];
    float* Wsl  = smem;                     // 128*512
    float* hbuf = smem + 128 * 512;         // 512
    float* gbuf = hbuf + 512;               // 128
    float* cbuf = gbuf + 128;               // 32

    float*    hglob = ws_state;             // 512
    float*    cglob = ws_state + 512;       // 512
    unsigned* bar   = (unsigned*)(ws_state + 1024);

    const int b   = (int)blockIdx.x;        // 0..15
    const int tid = (int)threadIdx.x;       // 0..255

#if HAVE_TDM
    // Stage W_hh slice with the Tensor Data Mover: 4 DMAs (one per gate's
    // 32 contiguous rows x 512 f32 = 64KB), issued by wave 0 only.
    if (tid < 32) {
        const unsigned lds_base = (unsigned)(size_t)Wsl;   // LDS byte address
        #pragma unroll
        for (int g = 0; g < 4; ++g) {
            const float* src = W_hh + ((size_t)g * H_DIM + (size_t)b * 32) * H_DIM;
            tdm_load_rows_f32(src, lds_base + (unsigned)g * 32u * H_DIM * 4u,
                              32u, (unsigned)H_DIM);
        }
        __builtin_amdgcn_s_wait_tensorcnt(0);
    }
#else
    // Fallback: VALU copy (float4, 256 floats per thread)
    for (int i = tid; i < (128 * 512) / 4; i += 256) {
        const int r = i >> 7;                   // local row 0..127 (= g*32+u)
        const int k = (i & 127) * 4;
        const int G = (r >> 5) * H_DIM + b * 32 + (r & 31);
        *(float4*)(Wsl + (size_t)r * H_DIM + k) =
            *(const float4*)(W_hh + (size_t)G * H_DIM + k);
    }
#endif
    if (tid < 32) cbuf[tid] = (c0 == 0) ? 0.0f : cglob[b * 32 + tid];
    __syncthreads();

    const int r  = tid >> 1;                // row 0..127
    const int k0 = (tid & 1) * 256;         // half of K
    const float* __restrict__ wr = Wsl + (size_t)r * H_DIM + k0;
    const int G = (r >> 5) * H_DIM + b * 32 + (r & 31);

    for (int t = c0; t < c0 + steps; ++t) {
        // pull current h into LDS; volatile -> bypass stale WGP$
        for (int i = tid; i < H_DIM; i += 256)
            hbuf[i] = ((volatile const float*)hglob)[i];
        __syncthreads();

        // prefetch next step's gates_x value (speculative, non-faulting)
        __builtin_prefetch(gates + (size_t)(t + 1 - c0) * G4H + G, 0, 1);

        float sum = 0.0f;
        const float* __restrict__ hr = hbuf + k0;
        #pragma unroll 8
        for (int k = 0; k < 256; k += 4) {
            float4 w  = *(const float4*)(wr + k);
            float4 h4 = *(const float4*)(hr + k);
            sum += w.x * h4.x + w.y * h4.y + w.z * h4.z + w.w * h4.w;
        }
        sum += __shfl_xor(sum, 1, 32);               // combine K halves (wave32)
        if ((tid & 1) == 0)
            gbuf[r] = gates[(size_t)(t - c0) * G4H + G] + sum;
        __syncthreads();

        if (tid < 32) {
            const float iv = sigmoidf_(gbuf[tid]);
            const float fv = sigmoidf_(gbuf[32 + tid]);
            const float gv = tanhf(gbuf[64 + tid]);
            const float ov = sigmoidf_(gbuf[96 + tid]);
            const float c  = fv * cbuf[tid] + iv * gv;
            cbuf[tid] = c;
            ((volatile float*)hglob)[b * 32 + tid] = ov * tanhf(c);
        }
        grid_barrier(bar, nblocks);
    }

    if (tid < 32) cglob[b * 32 + tid] = cbuf[tid];   // carry c to next chunk
}

// -------- Phase 3: log_softmax over a width-1 axis is identically 0 --------
__global__ void lstm_write_out(float* __restrict__ out, int n) {
    const int i = (int)blockIdx.x * 256 + (int)threadIdx.x;
    if (i < n) out[i] = 0.0f;
}

// ---------------------------------------------------------------------------
extern "C" void kernel_launch(void* const* d_in, const int* in_sizes, int n_in,
                              void* d_out, int out_size, void* d_ws, size_t ws_size,
                              hipStream_t stream) {
    (void)in_sizes; (void)n_in;
    const float* seq  = (const float*)d_in[0];
    const float* W_ih = (const float*)d_in[1];
    const float* W_hh = (const float*)d_in[2];
    const float* b_ih = (const float*)d_in[3];
    const float* b_hh = (const float*)d_in[4];
    // d_in[5]/d_in[6] (W_out, b_out) cannot affect the output: log_softmax
    // over a single-element axis is exactly zero.

    float* ws    = (float*)d_ws;
    float* gates = ws + WS_GATES_OFF / 4;

    // Size the gates staging chunk from available scratch (multiple of 32).
    size_t avail = (ws_size > WS_GATES_OFF) ? (ws_size - WS_GATES_OFF) : 0;
    long chunkT = (long)(avail / ((size_t)G4H * sizeof(float)));
    chunkT &= ~31L;
    if (chunkT < 32) chunkT = 32;
    if (chunkT > T_SEQ) chunkT = T_SEQ;

    const size_t smem_bytes = (size_t)(128 * 512 + 512 + 128 + 32) * sizeof(float);
    hipFuncSetAttribute((const void*)lstm_scan,
                        hipFuncAttributeMaxDynamicSharedMemorySize, (int)smem_bytes);

    lstm_init_state<<<1, 256, 0, stream>>>((unsigned*)d_ws);

    for (long c0 = 0; c0 < T_SEQ; c0 += chunkT) {
        const int steps = (int)(((T_SEQ - c0) < chunkT) ? (T_SEQ - c0) : chunkT);
        dim3 ggrid(G4H / 64, (unsigned)((steps + 31) / 32));
        lstm_gatesx_wmma<<<ggrid, 256, 0, stream>>>(seq, W_ih, b_ih, b_hh, gates, (int)c0);
        lstm_scan<<<16, 256, smem_bytes, stream>>>(W_hh, gates, ws, (int)c0, steps, 16u);
    }

    lstm_write_out<<<(out_size + 255) / 256, 256, 0, stream>>>((float*)d_out, out_size);
}